// GCN_63814624084109
// MI455X (gfx1250) — compile-verified
//
#include <hip/hip_runtime.h>
#include <math.h>

#define HD 128
#define BN_EPS 1e-5f
#define MT 4   // 16-row M-tiles per block -> 64 rows/block, B fragment reused 4x

typedef float v2f __attribute__((ext_vector_type(2)));
typedef float v8f __attribute__((ext_vector_type(8)));

// -------------------- utility: zero fill --------------------
__global__ __launch_bounds__(256) void k_zero(float* __restrict__ p, long long n) {
  long long i = (long long)blockIdx.x * blockDim.x + threadIdx.x;
  long long stride = (long long)gridDim.x * blockDim.x;
  for (; i < n; i += stride) p[i] = 0.0f;
}

// -------------------- degrees over row (src) and col (dst) --------------------
__global__ __launch_bounds__(256) void k_degree(const int* __restrict__ row, const int* __restrict__ col,
                                                float* __restrict__ degS, float* __restrict__ degC, int E) {
  int e = blockIdx.x * blockDim.x + threadIdx.x;
  if (e < E) {
    atomicAdd(&degS[row[e]], 1.0f);
    atomicAdd(&degC[col[e]], 1.0f);
  }
}

// deg -> dinv in place: deg>0 ? rsqrt(max(deg,1)) : 0
__global__ __launch_bounds__(256) void k_dinv(float* __restrict__ deg, int N) {
  int n = blockIdx.x * blockDim.x + threadIdx.x;
  if (n < N) {
    float d = deg[n];
    deg[n] = (d > 0.0f) ? rsqrtf(fmaxf(d, 1.0f)) : 0.0f;
  }
}

// x = concat(emb[N,127], log(deg_src+1)[:,None])
__global__ __launch_bounds__(128) void k_build_x(const float* __restrict__ emb, const float* __restrict__ degS,
                                                 float* __restrict__ x, int N) {
  int n = blockIdx.x;
  int f = threadIdx.x;
  if (n >= N) return;
  float v = (f < HD - 1) ? emb[(long long)n * (HD - 1) + f] : logf(degS[n] + 1.0f);
  x[(long long)n * HD + f] = v;
}

// norm[e] = dinv[row[e]] * dinv[col[e]]
__global__ __launch_bounds__(256) void k_norm(const int* __restrict__ row, const int* __restrict__ col,
                                              const float* __restrict__ dinv, float* __restrict__ nrm, int E) {
  int e = blockIdx.x * blockDim.x + threadIdx.x;
  if (e < E) nrm[e] = dinv[row[e]] * dinv[col[e]];
}

// -------------------- H = X @ W via V_WMMA_F32_16X16X4_F32 --------------------
// block = 256 threads = 8 waves covering 128 output columns; each wave computes
// MT stacked 16x16 tiles (64 rows per block). Per k-step one B fragment
// (W[k..k+3][n0..n0+15]) feeds MT WMMA ops -> 4x reuse of W, 128 wmma/wave.
// A layout (ISA 7.12.2, 32-bit A 16x4): lane l -> M = l%16; lanes<16 hold K=0,1;
// lanes>=16 hold K=2,3. B (4x16) mirrors with N = l%16. C/D: VGPR v -> row v (+8 hi).
__global__ __launch_bounds__(256) void k_gemm_wmma(const float* __restrict__ X, const float* __restrict__ W,
                                                   float* __restrict__ Hout, int N) {
  const int wave = threadIdx.x >> 5;
  const int lane = threadIdx.x & 31;
  const int lm = lane & 15;
  const int hi = lane >> 4;
  const int m0 = blockIdx.x * (16 * MT);
  const int n0 = wave * 16;

  const float* xrows[MT];
#pragma unroll
  for (int t = 0; t < MT; ++t) {
    int r = m0 + t * 16 + lm;
    int rc = (r < N) ? r : (N - 1);          // clamp: EXEC must stay all-ones for WMMA
    xrows[t] = X + (long long)rc * HD;
  }

  v8f acc[MT] = {};
#pragma unroll
  for (int k = 0; k < HD; k += 4) {
    const int ka = k + 2 * hi;               // even -> 8B-aligned float2 loads
    v2f b;
    b.x = W[ka * HD + n0 + lm];
    b.y = W[(ka + 1) * HD + n0 + lm];
#pragma unroll
    for (int t = 0; t < MT; ++t) {
      v2f a = *(const v2f*)(xrows[t] + ka);
      acc[t] = __builtin_amdgcn_wmma_f32_16x16x4_f32(false, a, false, b, (short)0, acc[t], false, false);
    }
  }

#pragma unroll
  for (int t = 0; t < MT; ++t) {
#pragma unroll
    for (int v = 0; v < 8; ++v) {
      int rr = m0 + t * 16 + v + 8 * hi;
      if (rr < N) Hout[(long long)rr * HD + n0 + lm] = acc[t][v];
    }
  }
}

// -------------------- agg[col] += norm * h[row], one wave per edge --------------------
__global__ __launch_bounds__(256) void k_scatter(const float* __restrict__ Hm, const int* __restrict__ row,
                                                 const int* __restrict__ col, const float* __restrict__ nrm,
                                                 float* __restrict__ agg, int E) {
  int e = blockIdx.x * 8 + (threadIdx.x >> 5);
  if (e >= E) return;
  int lane = threadIdx.x & 31;
  int r = row[e], c = col[e];
  float nm = nrm[e];
  float4 v = ((const float4*)(Hm + (long long)r * HD))[lane];   // coalesced 512B row gather
  float* dst = agg + (long long)c * HD + lane * 4;
  atomicAdd(dst + 0, nm * v.x);                                 // global_atomic_add_f32
  atomicAdd(dst + 1, nm * v.y);
  atomicAdd(dst + 2, nm * v.z);
  atomicAdd(dst + 3, nm * v.w);
}

// -------------------- per-feature sum / sumsq (thread = feature) --------------------
__global__ __launch_bounds__(128) void k_stats(const float* __restrict__ agg, float* __restrict__ st, int N) {
  int f = threadIdx.x;
  float s = 0.0f, s2 = 0.0f;
  for (int n = blockIdx.x; n < N; n += gridDim.x) {
    float v = agg[(long long)n * HD + f];
    s += v;
    s2 += v * v;
  }
  atomicAdd(&st[f], s);
  atomicAdd(&st[HD + f], s2);
}

// -------------------- BN (batch stats, biased var) + ReLU + optional residual ----------
// Reference adds bs[i] to agg before BN; with batch-statistic BN the bias cancels
// exactly in (agg - mean), so it is skipped.
__global__ __launch_bounds__(128) void k_bn(const float* __restrict__ agg, const float* __restrict__ st,
                                            const float* __restrict__ gamma, const float* __restrict__ beta,
                                            const float* __restrict__ xprev, float* __restrict__ xout,
                                            int N, int residual) {
  int f = threadIdx.x;
  float invN = 1.0f / (float)N;
  float mean = st[f] * invN;
  float var = st[HD + f] * invN - mean * mean;
  float inv = rsqrtf(var + BN_EPS);
  float g = gamma[f], b = beta[f];
  for (int n = blockIdx.x; n < N; n += gridDim.x) {
    long long idx = (long long)n * HD + f;
    float v = g * (agg[idx] - mean) * inv + b;
    v = fmaxf(v, 0.0f);
    if (residual) v += 0.5f * xprev[idx];
    xout[idx] = v;
  }
}

extern "C" void kernel_launch(void* const* d_in, const int* in_sizes, int n_in,
                              void* d_out, int out_size, void* d_ws, size_t ws_size,
                              hipStream_t stream) {
  (void)n_in; (void)out_size; (void)ws_size;
  const int H = HD;
  const float* emb    = (const float*)d_in[0];
  const float* Ws     = (const float*)d_in[1];
  // d_in[2] = bs: mathematically cancelled by batch-norm, unused.
  const float* gammas = (const float*)d_in[3];
  const float* betas  = (const float*)d_in[4];
  const int*   ei     = (const int*)d_in[5];

  const int N = in_sizes[0] / (H - 1);
  const int E = in_sizes[5] / 2;
  const int L = in_sizes[1] / (H * H);
  const int* row = ei;
  const int* col = ei + E;

  auto aln = [](size_t s) { return (s + 255) & ~(size_t)255; };
  char* p = (char*)d_ws;
  const size_t NB = (size_t)N * H * sizeof(float);
  float* x    = (float*)p; p += aln(NB);                 // node features (in-place across layers)
  float* h    = (float*)p; p += aln(NB);                 // x @ W
  float* agg  = (float*)p; p += aln(NB);                 // scattered messages
  float* degS = (float*)p; p += aln((size_t)N * 4);      // out-degree of row
  float* dinv = (float*)p; p += aln((size_t)N * 4);      // degC -> dinv in place
  float* nrm  = (float*)p; p += aln((size_t)E * 4);      // per-edge norm
  float* st   = (float*)p; p += aln((size_t)2 * H * 4);  // [sum(128), sumsq(128)]

  const int tb = 256;
  k_zero<<<1024, tb, 0, stream>>>(degS, N);
  k_zero<<<1024, tb, 0, stream>>>(dinv, N);
  k_degree<<<(E + tb - 1) / tb, tb, 0, stream>>>(row, col, degS, dinv, E);
  k_dinv<<<(N + tb - 1) / tb, tb, 0, stream>>>(dinv, N);
  k_build_x<<<N, H, 0, stream>>>(emb, degS, x, N);
  k_norm<<<(E + tb - 1) / tb, tb, 0, stream>>>(row, col, dinv, nrm, E);

  const int rows_per_block = 16 * MT;
  for (int l = 0; l < L; ++l) {
    const float* W = Ws + (size_t)l * H * H;
    k_gemm_wmma<<<(N + rows_per_block - 1) / rows_per_block, 256, 0, stream>>>(x, W, h, N);
    k_zero<<<4096, tb, 0, stream>>>(agg, (long long)N * H);
    k_scatter<<<(E + 7) / 8, 256, 0, stream>>>(h, row, col, nrm, agg, E);
    k_zero<<<1, 256, 0, stream>>>(st, 2 * H);
    k_stats<<<512, H, 0, stream>>>(agg, st, N);
    float* xout = (l == L - 1) ? (float*)d_out : x;   // same-index in-place is safe
    k_bn<<<1024, H, 0, stream>>>(agg, st, gammas + (size_t)l * H, betas + (size_t)l * H,
                                 x, xout, N, (l > 0) ? 1 : 0);
  }
}